// DMF_50130858279442
// MI455X (gfx1250) — compile-verified
//
#include <hip/hip_runtime.h>
#include <math.h>

// Problem constants (fixed by setup_inputs)
#define B_ROWS 1024
#define D_HALF 512
#define D_FULL 1024
#define NMUT   100
#define MRATE  0.2f
#define EPS_V  1e-6f

typedef __attribute__((ext_vector_type(2))) float v2f;
typedef __attribute__((ext_vector_type(8))) float v8f;

// ---------------------------------------------------------------------------
// threefry2x32-20 (JAX-family counter-based RNG): deterministic per (key,ctr)
// ---------------------------------------------------------------------------
__device__ __forceinline__ unsigned rotl32(unsigned x, unsigned n) {
  return (x << n) | (x >> (32u - n));
}

__device__ __forceinline__ void threefry2x32(unsigned k0, unsigned k1,
                                             unsigned c0, unsigned c1,
                                             unsigned& o0, unsigned& o1) {
  const unsigned ks2 = k0 ^ k1 ^ 0x1BD11BDAu;
  unsigned x0 = c0 + k0;
  unsigned x1 = c1 + k1;
#define TF_R(r) { x0 += x1; x1 = rotl32(x1, r); x1 ^= x0; }
  TF_R(13) TF_R(15) TF_R(26) TF_R(6)
  x0 += k1;  x1 += ks2 + 1u;
  TF_R(17) TF_R(29) TF_R(16) TF_R(24)
  x0 += ks2; x1 += k0 + 2u;
  TF_R(13) TF_R(15) TF_R(26) TF_R(6)
  x0 += k0;  x1 += k1 + 3u;
  TF_R(17) TF_R(29) TF_R(16) TF_R(24)
  x0 += k1;  x1 += ks2 + 4u;
  TF_R(13) TF_R(15) TF_R(26) TF_R(6)
  x0 += ks2; x1 += k0 + 5u;
#undef TF_R
  o0 = x0; o1 = x1;
}

// ---------------------------------------------------------------------------
// Kernel A: per-row std (ddof=1) for both tensors.
// Stages the 512-float row into LDS with CDNA5 async global->LDS copies.
// grid = 2048 rows, block = 128
// ---------------------------------------------------------------------------
__global__ __launch_bounds__(128) void dmf_row_std(const float* __restrict__ spatial,
                                                   const float* __restrict__ spectral,
                                                   float* __restrict__ std_out) {
  __shared__ float row[D_HALF];
  __shared__ float rs[128];
  __shared__ float rss[128];
  const int r   = blockIdx.x;   // 0..2047 (first 1024 = spatial)
  const int tid = threadIdx.x;  // 0..127
  const float* __restrict__ src = (r < B_ROWS)
      ? (spatial + (size_t)r * D_HALF)
      : (spectral + (size_t)(r - B_ROWS) * D_HALF);

  // Flat-pointer low 32 bits == LDS byte offset (hardware truncates to LDS addr)
  const unsigned lds_base = (unsigned)(uintptr_t)(&row[0]);
#pragma unroll
  for (int k = 0; k < 4; ++k) {
    const int idx = tid + k * 128;
    const unsigned lds_addr = lds_base + (unsigned)idx * 4u;
    const int goff = idx * 4;
    // GVS mode: reads global saddr (64-bit SGPR pair) + vaddr (32-bit offset), writes the LDS address in lds_addr
    asm volatile("global_load_async_to_lds_b32 %0, %1, %2"
                 :: "v"(lds_addr), "v"(goff), "s"(src)
                 : "memory");
  }
  asm volatile("s_wait_asynccnt 0" ::: "memory");
  __syncthreads();

  float s = 0.f, ss = 0.f;
#pragma unroll
  for (int k = 0; k < 4; ++k) {
    const float v = row[tid + k * 128];
    s += v; ss += v * v;
  }
  rs[tid] = s; rss[tid] = ss;
  __syncthreads();
  for (int off = 64; off > 0; off >>= 1) {
    if (tid < off) { rs[tid] += rs[tid + off]; rss[tid] += rss[tid + off]; }
    __syncthreads();
  }
  if (tid == 0) {
    const float n = (float)D_HALF;
    const float mean = rs[0] / n;
    const float var  = fmaxf((rss[0] - n * mean * mean) / (n - 1.0f), 0.0f);
    std_out[r] = sqrtf(var);
  }
}

// ---------------------------------------------------------------------------
// Kernel B: per output element, stream 100 mutation samples:
//   S1 = sum(delta), S2 = sum(delta^2), delta = mask * normal * std_row
// Emit w = 1/(var+eps) and pre = M*f + S1. Pure VALU (RNG-bound).
// Optimized: 2 mutations per pair of threefry calls -- one call supplies
// both mask uniforms (word0/word1), one call supplies both Box-Muller
// normals (r*cos and r*sin). Halves RNG work vs. 2 calls/mutation.
// grid = 4096, block = 256 (1M threads, one per [b,d])
// ---------------------------------------------------------------------------
__global__ __launch_bounds__(256) void dmf_stats(const float* __restrict__ spatial,
                                                 const float* __restrict__ spectral,
                                                 const float* __restrict__ std_arr,
                                                 float* __restrict__ w_out,
                                                 float* __restrict__ pre_out) {
  const int idx = blockIdx.x * 256 + threadIdx.x;    // 0 .. 1048575
  const int bi  = idx >> 10;                         // batch row
  const int d   = idx & (D_FULL - 1);                // concat column
  const int t   = d >> 9;                            // 0=spatial, 1=spectral
  const int dd  = d & (D_HALF - 1);

  const float f  = (t == 0) ? spatial[bi * D_HALF + dd] : spectral[bi * D_HALF + dd];
  const float sd = std_arr[t * B_ROWS + bi];

  // Four independent streams: {spatial,spectral} x {mask,noise}
  const unsigned km0 = 0x9E3779B9u * (unsigned)(2 * t + 1), km1 = 42u + (unsigned)(2 * t);
  const unsigned kn0 = 0x9E3779B9u * (unsigned)(2 * t + 2), kn1 = 43u + (unsigned)(2 * t);

  const float TO_U = 1.0f / 16777216.0f;             // 2^-24
  float S1 = 0.f, S2 = 0.f;
  for (int m = 0; m < NMUT; m += 2) {
    unsigned a0, a1, n0, n1;
    threefry2x32(km0, km1, (unsigned)idx, (unsigned)m, a0, a1);  // 2 mask uniforms
    threefry2x32(kn0, kn1, (unsigned)idx, (unsigned)m, n0, n1);  // 2 BM uniforms
    const float u1 = ((float)(n0 >> 8) + 0.5f) * TO_U;           // (0,1)
    const float u2 = (float)(n1 >> 8) * TO_U;                    // [0,1)
    const float rr = sqrtf(-2.0f * __logf(u1));
    float sn, cn;
    __sincosf(6.28318530718f * u2, &sn, &cn);
    const float nd0 = rr * cn * sd;                              // normal for m
    const float nd1 = rr * sn * sd;                              // normal for m+1
    const float d0 = ((float)(a0 >> 8) * TO_U < MRATE) ? nd0 : 0.0f;
    const float d1 = ((float)(a1 >> 8) * TO_U < MRATE) ? nd1 : 0.0f;
    S1 += d0 + d1;
    S2 += d0 * d0 + d1 * d1;
  }
  const float invM = 1.0f / (float)NMUT;
  const float m1   = S1 * invM;
  const float var  = fmaxf(S2 * invM - m1 * m1, 0.0f);
  w_out[idx]   = 1.0f / (var + EPS_V);
  pre_out[idx] = (float)NMUT * f + S1;
}

// ---------------------------------------------------------------------------
// Kernel C: colsum[d] = sum_b w[b,d] via V_WMMA_F32_16X16X4_F32.
// A = ones(16x4), B = 4x16 slab of w; accumulate 256 chained WMMAs.
// Fixed accumulation order -> deterministic (unlike float atomics).
// grid = 64 column tiles, block = 32 (one wave, EXEC all ones through WMMAs)
// ---------------------------------------------------------------------------
__global__ __launch_bounds__(32) void dmf_colsum_wmma(const float* __restrict__ w,
                                                      float* __restrict__ colsum) {
  const int lane = threadIdx.x;                 // 0..31
  const int col  = blockIdx.x * 16 + (lane & 15);
  const int koff = lane >> 4;                   // A/B layout: lanes 16-31 hold K+1 rows

  v2f a; a.x = 1.0f; a.y = 1.0f;                // 16x4 all-ones
  v8f c = {};
  const float* __restrict__ p0 = w + (size_t)koff * D_FULL + col;

#pragma unroll 4
  for (int k = 0; k < B_ROWS; k += 4) {
    v2f bm;                                     // B 4x16: VGPR0={K0,K1}, VGPR1={K2,K3}
    bm.x = p0[(size_t)k * D_FULL];
    bm.y = p0[(size_t)(k + 2) * D_FULL];
    c = __builtin_amdgcn_wmma_f32_16x16x4_f32(false, a, false, bm,
                                              (short)0, c, false, false);
  }
  // D layout: VGPR0, lanes 0-15 = row M=0, N=lane (all D rows equal here)
  if (lane < 16) colsum[col] = c[0];
}

// ---------------------------------------------------------------------------
// Kernel D: Z[b,d] = pre[b,d] * w[b,d] / colsum[d]
// ---------------------------------------------------------------------------
__global__ __launch_bounds__(256) void dmf_finalize(const float* __restrict__ w,
                                                    const float* __restrict__ pre,
                                                    const float* __restrict__ colsum,
                                                    float* __restrict__ out) {
  const int idx = blockIdx.x * 256 + threadIdx.x;
  const int d   = idx & (D_FULL - 1);
  out[idx] = pre[idx] * (w[idx] / colsum[d]);
}

// ---------------------------------------------------------------------------
extern "C" void kernel_launch(void* const* d_in, const int* in_sizes, int n_in,
                              void* d_out, int out_size, void* d_ws, size_t ws_size,
                              hipStream_t stream) {
  (void)in_sizes; (void)n_in; (void)out_size; (void)ws_size;
  const float* spatial  = (const float*)d_in[0];
  const float* spectral = (const float*)d_in[1];
  float* out = (float*)d_out;

  // Workspace layout (floats): std[2048] | w[1M] | pre[1M] | colsum[1024]
  float* ws       = (float*)d_ws;
  float* std_arr  = ws;
  float* w_buf    = ws + 2048;
  float* pre_buf  = w_buf + (size_t)B_ROWS * D_FULL;
  float* colsum   = pre_buf + (size_t)B_ROWS * D_FULL;

  dmf_row_std   <<<2 * B_ROWS, 128, 0, stream>>>(spatial, spectral, std_arr);
  dmf_stats     <<<(B_ROWS * D_FULL) / 256, 256, 0, stream>>>(spatial, spectral,
                                                              std_arr, w_buf, pre_buf);
  dmf_colsum_wmma<<<D_FULL / 16, 32, 0, stream>>>(w_buf, colsum);
  dmf_finalize  <<<(B_ROWS * D_FULL) / 256, 256, 0, stream>>>(w_buf, pre_buf,
                                                              colsum, out);
}